// Symmetric_Message_Network_90443421319354
// MI455X (gfx1250) — compile-verified
//
#include <hip/hip_runtime.h>

// Problem constants (HIDDEN fixed by the reference; N/E derived from in_sizes)
#define HIDDEN 128

typedef float v2f __attribute__((ext_vector_type(2)));
typedef float v8f __attribute__((ext_vector_type(8)));

// ---------------------------------------------------------------------------
// Kernel 1: zero the accumulator workspace (A [N x 128] and deg [N])
// ---------------------------------------------------------------------------
__global__ void smn_zero_kernel(float* __restrict__ p, long n) {
    long i = (long)blockIdx.x * blockDim.x + threadIdx.x;
    if (i < n) p[i] = 0.0f;
}

// ---------------------------------------------------------------------------
// Kernel 2: edge scatter. For each raw edge (s,d):
//   A[d] += x[s];  A[s] += x[d];  deg[d] += 1;  deg[s] += 1;
// One 32-lane wave slice per edge: lane handles 4 of 128 floats (float4).
// x and A live in L2 (25.6 MB each vs 192 MB L2) so atomics stay on-chip.
// ---------------------------------------------------------------------------
__global__ __launch_bounds__(256) void smn_edge_scatter(
    const float* __restrict__ x, const int* __restrict__ esrc,
    const int* __restrict__ edst, float* __restrict__ A,
    float* __restrict__ deg, int E) {
    int t = blockIdx.x * blockDim.x + threadIdx.x;
    int e = t >> 5;
    int lane = t & 31;
    if (e >= E) return;
    int s = esrc[e];
    int d = edst[e];
    float4 vs = *((const float4*)(x + (size_t)s * HIDDEN) + lane);
    float4 vd = *((const float4*)(x + (size_t)d * HIDDEN) + lane);
    float* Ad = A + (size_t)d * HIDDEN + lane * 4;
    float* As = A + (size_t)s * HIDDEN + lane * 4;
    atomicAdd(Ad + 0, vs.x); atomicAdd(Ad + 1, vs.y);
    atomicAdd(Ad + 2, vs.z); atomicAdd(Ad + 3, vs.w);
    atomicAdd(As + 0, vd.x); atomicAdd(As + 1, vd.y);
    atomicAdd(As + 2, vd.z); atomicAdd(As + 3, vd.w);
    if (lane == 0) {
        atomicAdd(deg + d, 1.0f);
        atomicAdd(deg + s, 1.0f);
    }
}

// ---------------------------------------------------------------------------
// Kernel 3: fused dual GEMM via V_WMMA_F32_16X16X4_F32.
//   out[v, :] = A[v, :] @ W1^T + (deg[v] * x[v, :]) @ W2^T
// W is [128, 256] row-major: W1[j,k] = W[j*256 + k], W2[j,k] = W[j*256 + 128 + k].
// One wave computes one 16x16 output tile; K=128 -> 32 steps x 2 WMMAs.
//
// Operand layouts (ISA 7.12.2, wave32):
//   A (16x4 f32, 2 VGPRs): lanes 0-15 -> M=lane, K={k0,k0+1};
//                          lanes 16-31 -> M=lane-16, K={k0+2,k0+3}
//   B (4x16, 2 VGPRs):     lanes 0-15 -> N=lane, K={k0,k0+1} (mirrored halves)
//                          B[k][n] = W[n][k] -> float2 load along W's row
//   C/D (8 VGPRs):         VGPR r: lanes 0-15 -> (M=r, N=lane);
//                          lanes 16-31 -> (M=r+8, N=lane-16)
// ---------------------------------------------------------------------------
__global__ __launch_bounds__(256) void smn_gemm_wmma(
    const float* __restrict__ A, const float* __restrict__ x,
    const float* __restrict__ deg, const float* __restrict__ W,
    float* __restrict__ out, int nvars) {
    const int wid  = threadIdx.x >> 5;          // wave in block: 0..7
    const int lane = threadIdx.x & 31;
    const int mt   = blockIdx.x;                // M tile (nvars/16 blocks)
    const int nt   = wid;                       // N tile: 0..7 (128/16)
    const int mbase = mt * 16;
    const int nbase = nt * 16;
    const int half  = lane >> 4;                // 0: K+{0,1}, 1: K+{2,3}
    const int l16   = lane & 15;

    int row = mbase + l16;                      // A-operand M row (both halves)
    if (row >= nvars) row = nvars - 1;          // clamp keeps EXEC all-1s
    const int wrow = nbase + l16;               // W row supplying B column n

    const float dscale = deg[row];

    const float* Arow  = A + (size_t)row * HIDDEN;
    const float* Xrow  = x + (size_t)row * HIDDEN;
    const float* Wrow1 = W + (size_t)wrow * (2 * HIDDEN);
    const float* Wrow2 = Wrow1 + HIDDEN;

    v8f c = {};
#pragma unroll 8
    for (int k0 = 0; k0 < HIDDEN; k0 += 4) {
        const int ka = k0 + half * 2;
        // GEMM 1: A_tile x W1^T
        v2f a1 = *(const v2f*)(Arow + ka);
        v2f b1 = *(const v2f*)(Wrow1 + ka);
        c = __builtin_amdgcn_wmma_f32_16x16x4_f32(
            false, a1, false, b1, (short)0, c, false, false);
        // GEMM 2: (deg * x)_tile x W2^T (fused accumulate)
        v2f a2 = *(const v2f*)(Xrow + ka);
        a2.x *= dscale;
        a2.y *= dscale;
        v2f b2 = *(const v2f*)(Wrow2 + ka);
        c = __builtin_amdgcn_wmma_f32_16x16x4_f32(
            false, a2, false, b2, (short)0, c, false, false);
    }

    // Store D: VGPR r holds rows {mbase+r (lanes 0-15), mbase+r+8 (lanes 16-31)}
    const int orow_base = mbase + half * 8;
#pragma unroll
    for (int r = 0; r < 8; ++r) {
        const int orow = orow_base + r;
        if (orow < nvars)
            out[(size_t)orow * HIDDEN + nbase + l16] = c[r];
    }
}

// ---------------------------------------------------------------------------
// Host launcher
// ---------------------------------------------------------------------------
extern "C" void kernel_launch(void* const* d_in, const int* in_sizes, int n_in,
                              void* d_out, int out_size, void* d_ws, size_t ws_size,
                              hipStream_t stream) {
    const float* x  = (const float*)d_in[0];             // [N, 128]
    const float* W  = (const float*)d_in[1];             // [128, 256]
    const int*   ei = (const int*)d_in[2];               // [2, E] flat

    const int N = in_sizes[0] / HIDDEN;                  // 50000
    const int E = in_sizes[2] / 2;                       // 400000
    const int* e0 = ei;                                  // src list
    const int* e1 = ei + E;                              // dst list

    // Workspace: A [N*128] floats, then deg [N] floats (~25.8 MB)
    float* A   = (float*)d_ws;
    float* deg = A + (size_t)N * HIDDEN;

    // 1) zero A + deg
    {
        long ztotal = (long)N * HIDDEN + N;
        int zblocks = (int)((ztotal + 255) / 256);
        smn_zero_kernel<<<zblocks, 256, 0, stream>>>(A, ztotal);
    }

    // 2) edge scatter: E edges * 32 lanes each, both directions per edge
    {
        long threads = (long)E * 32;
        int blocks = (int)((threads + 255) / 256);
        smn_edge_scatter<<<blocks, 256, 0, stream>>>(x, e0, e1, A, deg, E);
    }

    // 3) fused WMMA GEMM: one wave per 16x16 tile; N tiles=8 (one block's 8 waves)
    {
        int mtiles = (N + 15) / 16;                      // 3125
        smn_gemm_wmma<<<mtiles, 256, 0, stream>>>(A, x, deg, W,
                                                  (float*)d_out, N);
    }
}